// PointMLPStage_48180943127132
// MI455X (gfx1250) — compile-verified
//
#include <hip/hip_runtime.h>
#include <math.h>

typedef __attribute__((ext_vector_type(2))) float v2f;
typedef __attribute__((ext_vector_type(8))) float v8f;

#define BATCH 8
#define NPTS  2048
#define CIN   64
#define KNN   24
#define DIM   128
#define MOUT  1024

// W LDS layout: k-pair interleaved, padded. Pair row stride (floats):
#define WSTRIDE 288   // 128*2 data + 32 pad -> halves land on disjoint bank groups

// ---------------------------------------------------------------------------
// Kernel 1: brute-force kNN (K=24) per point. One thread = one query point.
// Batch xyz staged in LDS; top-K list kept sorted in LDS ([slot][thread]).
// ---------------------------------------------------------------------------
__global__ __launch_bounds__(256) void knn_kernel(const float* __restrict__ xyz,
                                                  int* __restrict__ knn_out) {
  extern __shared__ float smem[];
  float* sx = smem;                        // NPTS*3 floats
  float* sd = smem + NPTS * 3;             // KNN*256 floats
  int*   si = (int*)(sd + KNN * 256);      // KNN*256 ints
  const int tid = threadIdx.x;
  const int b   = blockIdx.y;
  const int n   = blockIdx.x * 256 + tid;

  for (int i = tid; i < NPTS * 3; i += 256)
    sx[i] = xyz[(size_t)b * NPTS * 3 + i];
  __syncthreads();

  const float qx = sx[n * 3 + 0], qy = sx[n * 3 + 1], qz = sx[n * 3 + 2];
#define DD(s) sd[(s)*256 + tid]
#define II(s) si[(s)*256 + tid]
  for (int s = 0; s < KNN; ++s) { DD(s) = 3.0e38f; II(s) = 0; }
  for (int j = 0; j < NPTS; ++j) {
    const float dx = sx[j * 3 + 0] - qx;
    const float dy = sx[j * 3 + 1] - qy;
    const float dz = sx[j * 3 + 2] - qz;
    const float d  = dx * dx + dy * dy + dz * dz;
    if (d < DD(KNN - 1)) {
      int pos = KNN - 1;
      while (pos > 0 && DD(pos - 1) > d) {
        DD(pos) = DD(pos - 1); II(pos) = II(pos - 1); --pos;
      }
      DD(pos) = d; II(pos) = j;
    }
  }
  int* dst = knn_out + ((size_t)b * NPTS + n) * KNN;
  for (int s = 0; s < KNN; ++s) dst[s] = II(s);
#undef DD
#undef II
}

// ---------------------------------------------------------------------------
// Per-wave 16x128 = (16x128) @ (128x128) GEMM using V_WMMA_F32_16X16X4_F32.
// A fragment (16x4 f32): lane holds A[lane&15][2*(lane>>4) + {0,1}]
// B fragment (4x16 f32): lane holds B[2*(lane>>4) + {0,1}][lane&15]
//   -> W stored k-pair interleaved in LDS so this is ONE aligned ds_load_b64.
// C/D (16x16 f32):       VGPR v, lane l -> C[v + 8*(l>>4)][l&15]
// ---------------------------------------------------------------------------
__device__ inline void wave_gemm_16x128(const float* __restrict__ Xl, int ld, int mbase,
                                        const float* __restrict__ Wl,
                                        float* __restrict__ Hl, int lane) {
  const int half = lane >> 4;
  const int l16  = lane & 15;
  v8f acc[8];
#pragma unroll
  for (int nt = 0; nt < 8; ++nt)
#pragma unroll
    for (int e = 0; e < 8; ++e) acc[nt][e] = 0.0f;

#pragma unroll 4
  for (int kt = 0; kt < 32; ++kt) {
    const int k0 = kt * 4 + half * 2;
    v2f a;
    a.x = Xl[(mbase + l16) * ld + k0];
    a.y = Xl[(mbase + l16) * ld + k0 + 1];
    const float* wrow = Wl + (kt * 2 + half) * WSTRIDE;
#pragma unroll
    for (int nt = 0; nt < 8; ++nt) {
      const v2f bf = *(const v2f*)(wrow + (nt * 16 + l16) * 2);
      acc[nt] = __builtin_amdgcn_wmma_f32_16x16x4_f32(
          false, a, false, bf, (short)0, acc[nt], false, false);
    }
  }
#pragma unroll
  for (int nt = 0; nt < 8; ++nt)
#pragma unroll
    for (int v = 0; v < 8; ++v)
      Hl[(mbase + v + 8 * half) * ld + nt * 16 + l16] = acc[nt][v];
}

// Stage a 128x128 fp32 weight matrix from global into LDS, k-pair interleaved:
//   Wl[(k>>1)*WSTRIDE + n*2 + (k&1)] = Wg[k*128 + n]
// Each iteration: two coalesced global b32 loads + one aligned ds_store_b64.
__device__ inline void stage_w(const float* __restrict__ Wg, float* __restrict__ Wl,
                               int tid, int nthr) {
  for (int i = tid; i < 64 * 128; i += nthr) {
    const int kp = i >> 7, n = i & 127;
    float2 v;
    v.x = Wg[(kp * 2)     * 128 + n];
    v.y = Wg[(kp * 2 + 1) * 128 + n];
    *(float2*)(Wl + kp * WSTRIDE + n * 2) = v;
  }
}

// Row-wise bias-add + LayerNorm(+optional ReLU) over 128 channels, in place.
__device__ inline void ln_rows(float* __restrict__ Hl, int ld, int rows,
                               const float* __restrict__ bias,
                               const float* __restrict__ gamma,
                               const float* __restrict__ betap,
                               bool relu, int tid, int nthr) {
  for (int r = tid; r < rows; r += nthr) {
    float* row = Hl + r * ld;
    float s = 0.f, s2 = 0.f;
    for (int c = 0; c < DIM; ++c) {
      const float v = row[c] + bias[c];
      s += v; s2 += v * v;
    }
    const float m    = s * (1.0f / DIM);
    const float var  = fmaxf(s2 * (1.0f / DIM) - m * m, 0.0f);
    const float rstd = rsqrtf(var + 1e-5f);
    for (int c = 0; c < DIM; ++c) {
      float v = (row[c] + bias[c] - m) * rstd * gamma[c] + betap[c];
      row[c] = relu ? fmaxf(v, 0.0f) : v;
    }
  }
}

// One residual block on a (rows x 128) LDS tile:
//   h = relu(LN(X@w1+b1)); h = LN(h@w2+b2); X = relu(h + X)
__device__ inline void res_block(float* Xl, float* Hl, float* H2l, float* Wl,
                                 int ld, int rows,
                                 const float* w1, const float* b1,
                                 const float* g1, const float* bb1,
                                 const float* w2, const float* b2,
                                 const float* g2, const float* bb2,
                                 int tid, int nthr) {
  const int wave = tid >> 5, lane = tid & 31, nwaves = nthr >> 5;
  const int mtiles = rows >> 4;

  stage_w(w1, Wl, tid, nthr);
  __syncthreads();
  for (int mt = wave; mt < mtiles; mt += nwaves)
    wave_gemm_16x128(Xl, ld, mt * 16, Wl, Hl, lane);
  __syncthreads();
  ln_rows(Hl, ld, rows, b1, g1, bb1, true, tid, nthr);
  __syncthreads();

  stage_w(w2, Wl, tid, nthr);
  __syncthreads();
  for (int mt = wave; mt < mtiles; mt += nwaves)
    wave_gemm_16x128(Hl, ld, mt * 16, Wl, H2l, lane);
  __syncthreads();
  ln_rows(H2l, ld, rows, b2, g2, bb2, false, tid, nthr);
  __syncthreads();

  for (int i = tid; i < rows * DIM; i += nthr) {
    const int r = i >> 7, c = i & (DIM - 1);
    Xl[r * ld + c] = fmaxf(H2l[r * ld + c] + Xl[r * ld + c], 0.0f);
  }
  __syncthreads();
}

// ---------------------------------------------------------------------------
// Kernel 2: fused gather + affine-normalize + 2 pre res-blocks + max-pool.
// WG = 4 points * 24 neighbors = 96 rows = 6 M-tiles = 6 waves (192 thr).
// ---------------------------------------------------------------------------
__global__ __launch_bounds__(192) void pre_kernel(
    const float* __restrict__ feat, const int* __restrict__ knn,
    const float* __restrict__ alpha, const float* __restrict__ beta,
    const float* __restrict__ w1, const float* __restrict__ b1,
    const float* __restrict__ g1, const float* __restrict__ bb1,
    const float* __restrict__ w2, const float* __restrict__ b2,
    const float* __restrict__ g2, const float* __restrict__ bb2,
    float* __restrict__ hpool) {
  extern __shared__ float smem[];
  const int ROWS = 96, LD = 129, P = 4;
  float* Xl  = smem;
  float* Hl  = Xl + ROWS * LD;
  float* H2l = Hl + ROWS * LD;
  float* Wl  = H2l + ROWS * LD;           // 64 * WSTRIDE
  const int tid = threadIdx.x, nthr = blockDim.x;
  const int pid0 = blockIdx.x * P;        // global point id base (same batch)

  // Gather K neighbors, geometric-affine normalize, build local tile in Xl:
  // cols [0,64) = center, cols [64,128) = normalized(grouped) - center.
  for (int t = tid; t < P * CIN; t += nthr) {
    const int p = t >> 6, c = t & 63;
    const int pid = pid0 + p;
    const size_t bbase = (size_t)(pid & ~(NPTS - 1));   // b * NPTS
    const int* nb = knn + (size_t)pid * KNN;
    const float center = feat[(size_t)pid * CIN + c];
    float s = 0.f, s2 = 0.f, vals[KNN];
#pragma unroll
    for (int k = 0; k < KNN; ++k) {
      const float v = feat[(bbase + (size_t)nb[k]) * CIN + c];
      vals[k] = v; s += v; s2 += v * v;
    }
    const float m   = s * (1.0f / KNN);
    const float var = fmaxf((s2 - (float)KNN * m * m) * (1.0f / (KNN - 1)), 0.0f);
    const float sd  = fmaxf(sqrtf(var), 1e-6f);
    const float al = alpha[c], be = beta[c];
#pragma unroll
    for (int k = 0; k < KNN; ++k) {
      const int r = p * KNN + k;
      const float gv = (vals[k] - m) / sd * al + be;
      Xl[r * LD + c]      = center;
      Xl[r * LD + 64 + c] = gv - center;
    }
  }
  __syncthreads();

  for (int blk = 0; blk < 2; ++blk) {
    const size_t wo = (size_t)blk * DIM * DIM, vo = (size_t)blk * DIM;
    res_block(Xl, Hl, H2l, Wl, LD, ROWS,
              w1 + wo, b1 + vo, g1 + vo, bb1 + vo,
              w2 + wo, b2 + vo, g2 + vo, bb2 + vo, tid, nthr);
  }

  // Max-pool over K neighbors (outputs are post-ReLU, so >= 0).
  for (int t = tid; t < P * DIM; t += nthr) {
    const int p = t >> 7, c = t & (DIM - 1);
    float mx = 0.0f;
    for (int k = 0; k < KNN; ++k) mx = fmaxf(mx, Xl[(p * KNN + k) * LD + c]);
    hpool[(size_t)(pid0 + p) * DIM + c] = mx;
  }
}

// ---------------------------------------------------------------------------
// Kernel 3: 2 pos res-blocks over pooled features, 128 rows / WG, in place.
// ---------------------------------------------------------------------------
__global__ __launch_bounds__(256) void pos_kernel(
    float* __restrict__ hpool,
    const float* __restrict__ w1, const float* __restrict__ b1,
    const float* __restrict__ g1, const float* __restrict__ bb1,
    const float* __restrict__ w2, const float* __restrict__ b2,
    const float* __restrict__ g2, const float* __restrict__ bb2) {
  extern __shared__ float smem[];
  const int ROWS = 128, LD = 129;
  float* Xl  = smem;
  float* Hl  = Xl + ROWS * LD;
  float* H2l = Hl + ROWS * LD;
  float* Wl  = H2l + ROWS * LD;
  const int tid = threadIdx.x, nthr = blockDim.x;
  const size_t row0 = (size_t)blockIdx.x * ROWS;

  for (int i = tid; i < ROWS * DIM; i += nthr) {
    const int r = i >> 7, c = i & (DIM - 1);
    Xl[r * LD + c] = hpool[(row0 + r) * DIM + c];
  }
  __syncthreads();

  for (int blk = 0; blk < 2; ++blk) {
    const size_t wo = (size_t)blk * DIM * DIM, vo = (size_t)blk * DIM;
    res_block(Xl, Hl, H2l, Wl, LD, ROWS,
              w1 + wo, b1 + vo, g1 + vo, bb1 + vo,
              w2 + wo, b2 + vo, g2 + vo, bb2 + vo, tid, nthr);
  }

  for (int i = tid; i < ROWS * DIM; i += nthr) {
    const int r = i >> 7, c = i & (DIM - 1);
    hpool[(row0 + r) * DIM + c] = Xl[r * LD + c];
  }
}

// ---------------------------------------------------------------------------
// Kernel 4/5: FPS gathers for the two outputs.
// ---------------------------------------------------------------------------
__global__ void gather_feat_kernel(const float* __restrict__ hpool,
                                   const int* __restrict__ fps,
                                   float* __restrict__ out) {
  const int i = blockIdx.x * blockDim.x + threadIdx.x;
  if (i >= BATCH * MOUT * DIM) return;
  const int c = i & (DIM - 1), t = i >> 7;
  const int b = t >> 10, m2 = t & (MOUT - 1);
  const int src = fps[b * MOUT + m2];
  out[i] = hpool[((size_t)b * NPTS + src) * DIM + c];
}

__global__ void gather_xyz_kernel(const float* __restrict__ xyz,
                                  const int* __restrict__ fps,
                                  float* __restrict__ out) {
  const int i = blockIdx.x * blockDim.x + threadIdx.x;
  if (i >= BATCH * MOUT * 3) return;
  const int d3 = i % 3, t = i / 3;
  const int b = t >> 10, m2 = t & (MOUT - 1);
  const int src = fps[b * MOUT + m2];
  out[i] = xyz[((size_t)b * NPTS + src) * 3 + d3];
}

// ---------------------------------------------------------------------------
extern "C" void kernel_launch(void* const* d_in, const int* in_sizes, int n_in,
                              void* d_out, int out_size, void* d_ws, size_t ws_size,
                              hipStream_t stream) {
  const float* feat  = (const float*)d_in[0];
  const float* xyz   = (const float*)d_in[1];
  const int*   fps   = (const int*)d_in[2];
  const float* alpha = (const float*)d_in[3];
  const float* beta  = (const float*)d_in[4];
  const float* pw1  = (const float*)d_in[5];
  const float* pb1  = (const float*)d_in[6];
  const float* pg1  = (const float*)d_in[7];
  const float* pbb1 = (const float*)d_in[8];
  const float* pw2  = (const float*)d_in[9];
  const float* pb2  = (const float*)d_in[10];
  const float* pg2  = (const float*)d_in[11];
  const float* pbb2 = (const float*)d_in[12];
  const float* qw1  = (const float*)d_in[13];
  const float* qb1  = (const float*)d_in[14];
  const float* qg1  = (const float*)d_in[15];
  const float* qbb1 = (const float*)d_in[16];
  const float* qw2  = (const float*)d_in[17];
  const float* qb2  = (const float*)d_in[18];
  const float* qg2  = (const float*)d_in[19];
  const float* qbb2 = (const float*)d_in[20];

  // workspace: [ knn idx : B*N*K ints ][ hpool : B*N*128 floats ]
  int*   knn_idx = (int*)d_ws;
  float* hpool   = (float*)((char*)d_ws + (size_t)BATCH * NPTS * KNN * sizeof(int));
  float* outF = (float*)d_out;
  float* outX = outF + (size_t)BATCH * MOUT * DIM;

  const size_t knn_lds = (size_t)NPTS * 3 * 4 + (size_t)KNN * 256 * 8;            // 73.7 KB
  const size_t pre_lds = (size_t)(3 * 96 * 129 + 64 * WSTRIDE) * 4;               // 222 KB
  const size_t pos_lds = (size_t)(3 * 128 * 129 + 64 * WSTRIDE) * 4;              // 272 KB
  hipFuncSetAttribute((const void*)knn_kernel, hipFuncAttributeMaxDynamicSharedMemorySize, (int)knn_lds);
  hipFuncSetAttribute((const void*)pre_kernel, hipFuncAttributeMaxDynamicSharedMemorySize, (int)pre_lds);
  hipFuncSetAttribute((const void*)pos_kernel, hipFuncAttributeMaxDynamicSharedMemorySize, (int)pos_lds);

  knn_kernel<<<dim3(NPTS / 256, BATCH), 256, knn_lds, stream>>>(xyz, knn_idx);

  pre_kernel<<<BATCH * NPTS / 4, 192, pre_lds, stream>>>(
      feat, knn_idx, alpha, beta,
      pw1, pb1, pg1, pbb1, pw2, pb2, pg2, pbb2, hpool);

  pos_kernel<<<BATCH * NPTS / 128, 256, pos_lds, stream>>>(
      hpool, qw1, qb1, qg1, qbb1, qw2, qb2, qg2, qbb2);

  gather_feat_kernel<<<(BATCH * MOUT * DIM + 255) / 256, 256, 0, stream>>>(hpool, fps, outF);
  gather_xyz_kernel<<<(BATCH * MOUT * 3 + 255) / 256, 256, 0, stream>>>(xyz, fps, outX);
}